// DifferentiableDroneController_29257317220585
// MI455X (gfx1250) — compile-verified
//
#include <hip/hip_runtime.h>
#include <cstdint>
#include <cstddef>

#define BLK 256

// gfx1250 async global->LDS copy path (ASYNCcnt-tracked). Guarded: if the
// builtins are absent on this toolchain we fall back to plain global loads.
#if defined(__HIP_DEVICE_COMPILE__) &&                                         \
    __has_builtin(__builtin_amdgcn_global_load_async_to_lds_b32) &&            \
    __has_builtin(__builtin_amdgcn_s_wait_asynccnt)
#define ASYNC_OK 1
typedef __attribute__((address_space(1))) int* as1ip;   // global src
typedef __attribute__((address_space(3))) int* as3ip;   // LDS dst
#if __has_builtin(__builtin_amdgcn_global_load_async_to_lds_b128)
#define ASYNC_B128 1
typedef int v4i_ __attribute__((ext_vector_type(4)));
typedef __attribute__((address_space(1))) v4i_* as1v4p;
typedef __attribute__((address_space(3))) v4i_* as3v4p;
#else
#define ASYNC_B128 0
#endif
#else
#define ASYNC_OK 0
#endif

__device__ __forceinline__ float clampf(float x, float lo, float hi) {
  return fminf(fmaxf(x, lo), hi);
}

#if ASYNC_OK
// Copy `words` floats from global `src` to LDS `dst` cooperatively
// (lane-contiguous, b128 wide path + b32 remainder).
__device__ __forceinline__ void tile_copy(const float* src, float* dst,
                                          int words, int tid) {
#if ASYNC_B128
  int q4 = words >> 2;  // number of 16B chunks
  for (int w = tid; w < q4; w += BLK)
    __builtin_amdgcn_global_load_async_to_lds_b128(
        (as1v4p)(src + 4 * w), (as3v4p)(dst + 4 * w), 0, 0);
  for (int w = (q4 << 2) + tid; w < words; w += BLK)
    __builtin_amdgcn_global_load_async_to_lds_b32(
        (as1ip)(src + w), (as3ip)(dst + w), 0, 0);
#else
  for (int w = tid; w < words; w += BLK)
    __builtin_amdgcn_global_load_async_to_lds_b32(
        (as1ip)(src + w), (as3ip)(dst + w), 0, 0);
#endif
}
#endif

// One evaluation of the drone rigid-body dynamics (mass = 1).
__device__ __forceinline__ void dyn(const float st[12], const float u[4],
                                    const float w[3], float k[12]) {
  const float Gc = 9.81f;
  const float IX = 0.01f, IY = 0.01f, IZ = 0.02f;
  const float INV_IX = 1.0f / IX, INV_IY = 1.0f / IY, INV_IZ = 1.0f / IZ;
  float vx = st[3], vy = st[4], vz = st[5];
  float c_r = __cosf(st[6]), s_r = __sinf(st[6]);
  float c_p = __cosf(st[7]), s_p = __sinf(st[7]);
  float c_y = __cosf(st[8]), s_y = __sinf(st[8]);
  float wx = st[9], wy = st[10], wz = st[11];

  float T = u[0];
  float tgx = T * (c_y * s_p * c_r + s_y * s_r);
  float tgy = T * (s_y * s_p * c_r - c_y * s_r);
  float tgz = T * (c_p * c_r);

  float rx = vx - w[0], ry = vy - w[1], rz = vz - w[2];
  float vn = sqrtf(rx * rx + ry * ry + rz * rz);
  float ax = tgx + (-0.05f * vn * rx);
  float ay = tgy + (-0.05f * vn * ry);
  float az = tgz + (-0.05f * vn * rz) - Gc;

  // omega x (I*omega)
  float bx = IX * wx, by = IY * wy, bz = IZ * wz;
  float cx = wy * bz - wz * by;
  float cy = wz * bx - wx * bz;
  float cz = wx * by - wy * bx;
  float odx = (u[1] - cx) * INV_IX;
  float ody = (u[2] - cy) * INV_IY;
  float odz = (u[3] - cz) * INV_IZ;

  float cp = c_p;
  if (fabsf(cp) < 1e-6f) cp = (cp > 0.0f) ? 1e-6f : ((cp < 0.0f) ? -1e-6f : 1e-6f);
  float rcp_cp = 1.0f / cp;  // one IEEE div, reused 3x
  float tpc = s_p * rcp_cp;
  float roll_dot  = wx + s_r * tpc * wy + c_r * tpc * wz;
  float pitch_dot = c_r * wy - s_r * wz;
  float yaw_dot   = (s_r * rcp_cp) * wy + (c_r * rcp_cp) * wz;

  k[0] = vx;       k[1] = vy;        k[2] = vz;
  k[3] = ax;       k[4] = ay;        k[5] = az;
  k[6] = roll_dot; k[7] = pitch_dot; k[8] = yaw_dot;
  k[9] = odx;      k[10] = ody;      k[11] = odz;
}

__global__ void __launch_bounds__(BLK)
drone_step(const float* __restrict__ state, const float* __restrict__ tpos,
           const float* __restrict__ tyaw, const float* __restrict__ wind,
           const float* __restrict__ pverr, const float* __restrict__ prerr,
           const float* __restrict__ ipos, const float* __restrict__ ivel,
           const float* __restrict__ irate,
           const float* __restrict__ kp_pos, const float* __restrict__ ki_pos,
           const float* __restrict__ kp_vel, const float* __restrict__ ki_vel,
           const float* __restrict__ kd_vel, const float* __restrict__ kp_att,
           const float* __restrict__ kp_rate, const float* __restrict__ ki_rate,
           const float* __restrict__ kd_rate,
           float* __restrict__ out, int n) {
  const float DTc = 0.01f;
  const float INV_DT = 1.0f / DTc;   // compile-time folded, correctly rounded
  const float Gc = 9.81f;
  const float PI_F   = 3.14159265358979323846f;
  const float TWO_PI = 6.28318530717958647692f;

  int tid = threadIdx.x;
  long long base = (long long)blockIdx.x * BLK;
  long long i = base + tid;

#if ASYNC_OK
  // Stage the whole 256-drone tile into LDS with wide async global->LDS DMA.
  __shared__ __align__(16) float ls_state[BLK * 12];
  __shared__ __align__(16) float ls3[7][BLK * 3];
  __shared__ __align__(16) float ls_ty[BLK];
  {
    int tile_n = n - (int)base;
    if (tile_n > BLK) tile_n = BLK;
    tile_copy(state + base * 12, ls_state, tile_n * 12, tid);
    const float* g3[7] = {tpos + base * 3,  wind + base * 3, pverr + base * 3,
                          prerr + base * 3, ipos + base * 3, ivel + base * 3,
                          irate + base * 3};
#pragma unroll
    for (int a = 0; a < 7; ++a) tile_copy(g3[a], ls3[a], tile_n * 3, tid);
    tile_copy(tyaw + base, ls_ty, tile_n, tid);
    __builtin_amdgcn_s_wait_asynccnt(0);
    __syncthreads();
  }
#endif

  if (i >= n) return;

  float st[12], tp3[3], wv[3], pv[3], pr[3], ip3[3], iv3[3], ir3[3], ty;
#if ASYNC_OK
#pragma unroll
  for (int j = 0; j < 12; ++j) st[j] = ls_state[tid * 12 + j];  // 3x ds_load_b128
#pragma unroll
  for (int c = 0; c < 3; ++c) {
    tp3[c] = ls3[0][tid * 3 + c];
    wv[c]  = ls3[1][tid * 3 + c];
    pv[c]  = ls3[2][tid * 3 + c];
    pr[c]  = ls3[3][tid * 3 + c];
    ip3[c] = ls3[4][tid * 3 + c];
    iv3[c] = ls3[5][tid * 3 + c];
    ir3[c] = ls3[6][tid * 3 + c];
  }
  ty = ls_ty[tid];
#else
  {
    const float4* sp = (const float4*)(state + i * 12);  // 48B rows: aligned
    float4 a0 = sp[0], a1 = sp[1], a2 = sp[2];
    st[0] = a0.x; st[1] = a0.y; st[2]  = a0.z; st[3]  = a0.w;
    st[4] = a1.x; st[5] = a1.y; st[6]  = a1.z; st[7]  = a1.w;
    st[8] = a2.x; st[9] = a2.y; st[10] = a2.z; st[11] = a2.w;
#pragma unroll
    for (int c = 0; c < 3; ++c) {
      tp3[c] = tpos[i * 3 + c];
      wv[c]  = wind[i * 3 + c];
      pv[c]  = pverr[i * 3 + c];
      pr[c]  = prerr[i * 3 + c];
      ip3[c] = ipos[i * 3 + c];
      iv3[c] = ivel[i * 3 + c];
      ir3[c] = irate[i * 3 + c];
    }
    ty = tyaw[i];
  }
#endif

  // Gains are wave-uniform -> scalar loads.
  float akpp[3], akip[3], akpv[3], akiv[3], akdv[3], akpa[3], akpr[3], akir[3],
      akdr[3];
#pragma unroll
  for (int c = 0; c < 3; ++c) {
    akpp[c] = fabsf(kp_pos[c]);
    akip[c] = fabsf(ki_pos[c]);
    akpv[c] = fabsf(kp_vel[c]);
    akiv[c] = fabsf(ki_vel[c]);
    akdv[c] = fabsf(kd_vel[c]);
    akpa[c] = fabsf(kp_att[c]);
    akpr[c] = fabsf(kp_rate[c]);
    akir[c] = fabsf(ki_rate[c]);
    akdr[c] = fabsf(kd_rate[c]);
  }

  // ---------------- controller ----------------
  float des_acc[3];
#pragma unroll
  for (int c = 0; c < 3; ++c) {
    float pe  = tp3[c] - st[c];
    float ipn = clampf(ip3[c] + pe * DTc, -2.0f, 2.0f);
    float dv  = clampf(pe * akpp[c] + ipn * akip[c], -10.0f, 10.0f);
    float ve  = dv - st[3 + c];
    float ivn = clampf(iv3[c] + ve * DTc, -2.0f, 2.0f);
    float vd  = (ve - pv[c]) * INV_DT;
    des_acc[c] = ve * akpv[c] + ivn * akiv[c] + vd * akdv[c];
  }
  float an = sqrtf(des_acc[0] * des_acc[0] + des_acc[1] * des_acc[1] +
                   des_acc[2] * des_acc[2]);
  bool big = an > 10.0f;
  float q10 = 10.0f / an;
#pragma unroll
  for (int c = 0; c < 3; ++c) {
    float lim = des_acc[c] * q10;
    des_acc[c] = big ? lim : des_acc[c];
  }
  float tvx = des_acc[0], tvy = des_acc[1], tvz = des_acc[2] + Gc;
  float tvn = sqrtf(tvx * tvx + tvy * tvy + tvz * tvz);
  float Tthr = clampf(tvn, (float)(0.1 * 9.81), (float)(2.0 * 9.81));
  float rtn = 1.0f / tvn;
  float tnx = tvx * rtn, tny = tvy * rtn;

  float s_ty = __sinf(ty), c_ty = __cosf(ty);
  float roll_arg = clampf(tnx * s_ty - tny * c_ty, -0.999f, 0.999f);
  float des_roll = asinf(roll_arg);
  float pitch_arg =
      clampf((tnx * c_ty + tny * s_ty) / __cosf(des_roll), -0.999f, 0.999f);
  float des_pitch = asinf(pitch_arg);
  des_roll  = clampf(des_roll, -0.523f, 0.523f);
  des_pitch = clampf(des_pitch, -0.523f, 0.523f);

  float ae[3];
  ae[0] = des_roll - st[6];
  ae[1] = des_pitch - st[7];
  {
    float a2 = ty - st[8] + PI_F;           // jnp.mod(x + pi, 2pi) - pi
    a2 = a2 - floorf(a2 / TWO_PI) * TWO_PI;
    ae[2] = a2 - PI_F;
  }

  float u[4];
  u[0] = Tthr;
  const float IVEC[3] = {0.01f, 0.01f, 0.02f};
#pragma unroll
  for (int c = 0; c < 3; ++c) {
    float dr  = ae[c] * akpa[c];
    float re  = dr - st[9 + c];
    float irn = clampf(ir3[c] + re * DTc, -1.0f, 1.0f);
    float rd  = (re - pr[c]) * INV_DT;
    float dtq = re * akpr[c] + irn * akir[c] + rd * akdr[c];
    u[1 + c] = dtq * IVEC[c];
  }

  // ---------------- RK4 ----------------
  float k[12], accum[12], tmp[12];
  dyn(st, u, wv, k);
#pragma unroll
  for (int j = 0; j < 12; ++j) { accum[j] = k[j]; tmp[j] = st[j] + 0.005f * k[j]; }
  dyn(tmp, u, wv, k);
#pragma unroll
  for (int j = 0; j < 12; ++j) { accum[j] += 2.0f * k[j]; tmp[j] = st[j] + 0.005f * k[j]; }
  dyn(tmp, u, wv, k);
#pragma unroll
  for (int j = 0; j < 12; ++j) { accum[j] += 2.0f * k[j]; tmp[j] = st[j] + 0.01f * k[j]; }
  dyn(tmp, u, wv, k);

  const float DT6 = (float)(0.01 / 6.0);
  float o[12];
#pragma unroll
  for (int j = 0; j < 12; ++j) {
    accum[j] += k[j];
    o[j] = st[j] + DT6 * accum[j];
  }

  float4* op = (float4*)(out + i * 12);  // 48B rows: aligned b128 stores
  op[0] = make_float4(o[0], o[1], o[2],  o[3]);
  op[1] = make_float4(o[4], o[5], o[6],  o[7]);
  op[2] = make_float4(o[8], o[9], o[10], o[11]);
}

extern "C" void kernel_launch(void* const* d_in, const int* in_sizes, int n_in,
                              void* d_out, int out_size, void* d_ws,
                              size_t ws_size, hipStream_t stream) {
  (void)n_in; (void)out_size; (void)d_ws; (void)ws_size;
  int n = in_sizes[0] / 12;  // state is [B, 12]
  dim3 grid((unsigned)((n + BLK - 1) / BLK)), block(BLK);
  drone_step<<<grid, block, 0, stream>>>(
      (const float*)d_in[0], (const float*)d_in[1], (const float*)d_in[2],
      (const float*)d_in[3], (const float*)d_in[4], (const float*)d_in[5],
      (const float*)d_in[6], (const float*)d_in[7], (const float*)d_in[8],
      (const float*)d_in[9], (const float*)d_in[10], (const float*)d_in[11],
      (const float*)d_in[12], (const float*)d_in[13], (const float*)d_in[14],
      (const float*)d_in[15], (const float*)d_in[16], (const float*)d_in[17],
      (float*)d_out, n);
}